// AttnDecoderRNN_57535381897455
// MI455X (gfx1250) — compile-verified
//
#include <hip/hip_runtime.h>
#include <math.h>

#define H   1024
#define V   50257
#define L   512
#define H2  2048
#define H3  3072

typedef __attribute__((ext_vector_type(2))) float v2f;
typedef __attribute__((ext_vector_type(8))) float v8f;

__device__ __forceinline__ float wave_reduce_sum(float s) {
#pragma unroll
    for (int ofs = 16; ofs > 0; ofs >>= 1) s += __shfl_xor(s, ofs, 32);
    return s;
}

// ---------------------------------------------------------------------------
// K1: cat1 = [embedded | h0]   (2048 floats)
// ---------------------------------------------------------------------------
__global__ void k_cat1(const int* __restrict__ tok,
                       const float* __restrict__ emb,
                       const float* __restrict__ hidden,
                       float* __restrict__ cat1) {
    int i = blockIdx.x * blockDim.x + threadIdx.x;
    if (i < H2) cat1[i] = (i < H) ? emb[(size_t)tok[0] * H + i] : hidden[i - H];
}

// ---------------------------------------------------------------------------
// K2: attn logits[l] = dot(cat1, attn_W[l,:]) + attn_b[l]; one wave per row
// ---------------------------------------------------------------------------
__global__ void __launch_bounds__(256)
k_attn_scores(const float* __restrict__ cat1,
              const float* __restrict__ attn_W,
              const float* __restrict__ attn_b,
              float* __restrict__ logits) {
    int wave = (blockIdx.x * 256 + threadIdx.x) >> 5;   // 0..511
    int lane = threadIdx.x & 31;
    const float4* w4 = (const float4*)(attn_W + (size_t)wave * H2);
    const float4* v4 = (const float4*)cat1;
    float s = 0.0f;
#pragma unroll
    for (int i = 0; i < 16; ++i) {
        float4 a = v4[i * 32 + lane];
        float4 b = w4[i * 32 + lane];
        s += a.x * b.x + a.y * b.y + a.z * b.z + a.w * b.w;
    }
    s = wave_reduce_sum(s);
    if (lane == 0) logits[wave] = s + attn_b[wave];
}

// ---------------------------------------------------------------------------
// K3: softmax over 512 logits -> attn_weights (d_out tail)
// ---------------------------------------------------------------------------
__global__ void __launch_bounds__(512)
k_softmax512(const float* __restrict__ logits, float* __restrict__ weights) {
    __shared__ float red[512];
    int t = threadIdx.x;
    float v = logits[t];
    red[t] = v;
    __syncthreads();
    for (int ofs = 256; ofs > 0; ofs >>= 1) {
        if (t < ofs) red[t] = fmaxf(red[t], red[t + ofs]);
        __syncthreads();
    }
    float m = red[0];
    __syncthreads();
    float e = __expf(v - m);
    red[t] = e;
    __syncthreads();
    for (int ofs = 256; ofs > 0; ofs >>= 1) {
        if (t < ofs) red[t] += red[t + ofs];
        __syncthreads();
    }
    weights[t] = e / red[0];
}

// ---------------------------------------------------------------------------
// K4a: partial context: part[p][h] = sum_{l in p-th 64-row slab} w[l]*enc[l,h]
//      grid (4, 8) x 256 threads. K4b combines the 8 slabs.
// ---------------------------------------------------------------------------
__global__ void __launch_bounds__(256)
k_attn_partial(const float* __restrict__ w,
               const float* __restrict__ enc,
               float* __restrict__ part) {
    int h  = blockIdx.x * 256 + threadIdx.x;
    int l0 = blockIdx.y * (L / 8);
    float acc = 0.0f;
#pragma unroll 4
    for (int l = l0; l < l0 + L / 8; ++l) acc += w[l] * enc[(size_t)l * H + h];
    part[(size_t)blockIdx.y * H + h] = acc;
}

__global__ void k_attn_combine(const float* __restrict__ part,
                               float* __restrict__ applied) {
    int h = blockIdx.x * 256 + threadIdx.x;
    float acc = 0.0f;
#pragma unroll
    for (int p = 0; p < 8; ++p) acc += part[(size_t)p * H + h];
    applied[h] = acc;
}

// ---------------------------------------------------------------------------
// K5: x[r] = relu(dot([embedded|applied], comb_W[r,:]) + comb_b[r]); wave/row
// ---------------------------------------------------------------------------
__global__ void __launch_bounds__(256)
k_combine(const float* __restrict__ cat1,
          const float* __restrict__ applied,
          const float* __restrict__ comb_W,
          const float* __restrict__ comb_b,
          float* __restrict__ xbuf) {
    int row  = (blockIdx.x * 256 + threadIdx.x) >> 5;   // 0..1023
    int lane = threadIdx.x & 31;
    const float4* w4 = (const float4*)(comb_W + (size_t)row * H2);
    const float4* e4 = (const float4*)cat1;     // first H = embedded
    const float4* a4 = (const float4*)applied;
    float s = 0.0f;
#pragma unroll
    for (int i = 0; i < 8; ++i) {               // j in [0, H): embedded part
        float4 a = e4[i * 32 + lane];
        float4 b = w4[i * 32 + lane];
        s += a.x * b.x + a.y * b.y + a.z * b.z + a.w * b.w;
    }
#pragma unroll
    for (int i = 0; i < 8; ++i) {               // j in [H, 2H): attn_applied
        float4 a = a4[i * 32 + lane];
        float4 b = w4[(i + 8) * 32 + lane];
        s += a.x * b.x + a.y * b.y + a.z * b.z + a.w * b.w;
    }
    s = wave_reduce_sum(s);
    if (lane == 0) xbuf[row] = fmaxf(s + comb_b[row], 0.0f);
}

// ---------------------------------------------------------------------------
// K6: GRU gate GEMVs; one wave per output row (6144 waves).
//     waves 0..3071 -> gi = Wih@x + bih ; 3072..6143 -> gh = Whh@h0 + bhh
// ---------------------------------------------------------------------------
__global__ void __launch_bounds__(256)
k_gru_gemv(const float* __restrict__ xbuf,
           const float* __restrict__ hidden,
           const float* __restrict__ Wih,
           const float* __restrict__ Whh,
           const float* __restrict__ bih,
           const float* __restrict__ bhh,
           float* __restrict__ gi,
           float* __restrict__ gh) {
    int gw   = (blockIdx.x * 256 + threadIdx.x) >> 5;   // 0..6143
    int lane = threadIdx.x & 31;
    const float4* v4;
    const float4* w4;
    float bias;
    float* outp;
    int r;
    if (gw < H3) { r = gw;      v4 = (const float4*)xbuf;   w4 = (const float4*)(Wih + (size_t)r * H); bias = bih[r]; outp = gi; }
    else         { r = gw - H3; v4 = (const float4*)hidden; w4 = (const float4*)(Whh + (size_t)r * H); bias = bhh[r]; outp = gh; }
    float s = 0.0f;
#pragma unroll
    for (int i = 0; i < 8; ++i) {
        float4 a = v4[i * 32 + lane];
        float4 b = w4[i * 32 + lane];
        s += a.x * b.x + a.y * b.y + a.z * b.z + a.w * b.w;
    }
    s = wave_reduce_sum(s);
    if (lane == 0) outp[r] = s + bias;
}

// ---------------------------------------------------------------------------
// K7: GRU elementwise update -> h_new (d_out[V : V+H])
// ---------------------------------------------------------------------------
__global__ void k_gru_update(const float* __restrict__ gi,
                             const float* __restrict__ gh,
                             const float* __restrict__ hidden,
                             float* __restrict__ hnew) {
    int k = blockIdx.x * blockDim.x + threadIdx.x;
    if (k >= H) return;
    float r = 1.0f / (1.0f + __expf(-(gi[k] + gh[k])));
    float z = 1.0f / (1.0f + __expf(-(gi[k + H] + gh[k + H])));
    float n = tanhf(gi[k + 2 * H] + r * gh[k + 2 * H]);
    hnew[k] = (1.0f - z) * n + z * hidden[k];
}

// ---------------------------------------------------------------------------
// K8: vocab projection via V_WMMA_F32_16X16X4_F32, two independent
//     accumulator chains so back-to-back WMMAs pipeline.
//     A: row 0 = h_new K-chunk (lane0: K=0,1; lane16: K=2,3), rows 1-15 zero.
//     B: lane n   (n<16) = out_W[v0+n, k0+{0,1}]
//        lane n+16       = out_W[v0+n, k0+{2,3}]
//     D[0][n] lands in c[0] of lanes 0..15 -> 16 logits per wave.
//     EXEC all-1s: masking by select only, no divergence around WMMA.
// ---------------------------------------------------------------------------
__global__ void __launch_bounds__(256)
k_out_proj_wmma(const float* __restrict__ hvec,
                const float* __restrict__ out_W,
                const float* __restrict__ out_b,
                float* __restrict__ logits) {
    int wave = (blockIdx.x * 256 + threadIdx.x) >> 5;
    int lane = threadIdx.x & 31;
    int half = lane >> 4;          // 0: K=0,1   1: K=2,3
    int lrow = lane & 15;          // B column N / A row M
    int v0 = wave * 16;

    int vrow = v0 + lrow;
    if (vrow > V - 1) vrow = V - 1;             // clamp; store masked below
    const float* wr = out_W + (size_t)vrow * H + (half << 1);
    const float* xr = hvec + (half << 1);
    bool isrow0 = (lrow == 0);

    v8f c0 = {0.0f, 0.0f, 0.0f, 0.0f, 0.0f, 0.0f, 0.0f, 0.0f};
    v8f c1 = {0.0f, 0.0f, 0.0f, 0.0f, 0.0f, 0.0f, 0.0f, 0.0f};

#pragma unroll 2
    for (int k0 = 0; k0 < H; k0 += 8) {
        float2 wv0 = *(const float2*)(wr + k0);
        float2 xv0 = *(const float2*)(xr + k0);
        v2f b0;  b0[0] = wv0.x;  b0[1] = wv0.y;
        v2f a0;  a0[0] = isrow0 ? xv0.x : 0.0f;  a0[1] = isrow0 ? xv0.y : 0.0f;
        c0 = __builtin_amdgcn_wmma_f32_16x16x4_f32(false, a0, false, b0,
                                                   (short)0, c0, false, false);

        float2 wv1 = *(const float2*)(wr + k0 + 4);
        float2 xv1 = *(const float2*)(xr + k0 + 4);
        v2f b1;  b1[0] = wv1.x;  b1[1] = wv1.y;
        v2f a1;  a1[0] = isrow0 ? xv1.x : 0.0f;  a1[1] = isrow0 ? xv1.y : 0.0f;
        c1 = __builtin_amdgcn_wmma_f32_16x16x4_f32(false, a1, false, b1,
                                                   (short)0, c1, false, false);
    }

    int vout = v0 + lane;
    if (lane < 16 && vout < V) logits[vout] = c0[0] + c1[0] + out_b[vout];
}

// ---------------------------------------------------------------------------
// K9: grid-parallel log-softmax. (a) 64-block online (max,sum) partials,
//     (b) tiny combine -> lse, (c) grid-parallel subtract.
// ---------------------------------------------------------------------------
#define NPB 64   // partial blocks

__global__ void __launch_bounds__(256)
k_lse_partial(const float* __restrict__ logits,
              float* __restrict__ pm, float* __restrict__ ps) {
    __shared__ float sm[256], ss[256];
    int t = threadIdx.x;
    int g = blockIdx.x * 256 + t;
    float m = -INFINITY, s = 0.0f;
    for (int v = g; v < V; v += NPB * 256) {
        float x = logits[v];
        float nm = fmaxf(m, x);
        s = s * __expf(m - nm) + __expf(x - nm);
        m = nm;
    }
    sm[t] = m; ss[t] = s;
    __syncthreads();
    for (int ofs = 128; ofs > 0; ofs >>= 1) {
        if (t < ofs) {
            float M = fmaxf(sm[t], sm[t + ofs]);
            float S = ss[t] * __expf(sm[t] - M) + ss[t + ofs] * __expf(sm[t + ofs] - M);
            sm[t] = M; ss[t] = S;
        }
        __syncthreads();
    }
    if (t == 0) { pm[blockIdx.x] = sm[0]; ps[blockIdx.x] = ss[0]; }
}

__global__ void __launch_bounds__(NPB)
k_lse_final(const float* __restrict__ pm, const float* __restrict__ ps,
            float* __restrict__ lse) {
    __shared__ float sm[NPB], ss[NPB];
    int t = threadIdx.x;
    sm[t] = pm[t]; ss[t] = ps[t];
    __syncthreads();
    for (int ofs = NPB / 2; ofs > 0; ofs >>= 1) {
        if (t < ofs) {
            float M = fmaxf(sm[t], sm[t + ofs]);
            float S = ss[t] * __expf(sm[t] - M) + ss[t + ofs] * __expf(sm[t + ofs] - M);
            sm[t] = M; ss[t] = S;
        }
        __syncthreads();
    }
    if (t == 0) lse[0] = sm[0] + __logf(ss[0]);
}

__global__ void __launch_bounds__(256)
k_lsm_apply(float* __restrict__ logits, const float* __restrict__ lse) {
    float l = lse[0];
    int g = blockIdx.x * 256 + threadIdx.x;
    for (int v = g; v < V; v += NPB * 256) logits[v] -= l;
}

// ---------------------------------------------------------------------------
// launch
// ---------------------------------------------------------------------------
extern "C" void kernel_launch(void* const* d_in, const int* in_sizes, int n_in,
                              void* d_out, int out_size, void* d_ws, size_t ws_size,
                              hipStream_t stream) {
    const int*   tok     = (const int*)  d_in[0];
    const float* hidden  = (const float*)d_in[1];
    const float* enc     = (const float*)d_in[2];
    const float* emb     = (const float*)d_in[3];
    const float* attn_W  = (const float*)d_in[4];
    const float* attn_b  = (const float*)d_in[5];
    const float* comb_W  = (const float*)d_in[6];
    const float* comb_b  = (const float*)d_in[7];
    const float* Wih     = (const float*)d_in[8];
    const float* Whh     = (const float*)d_in[9];
    const float* bih     = (const float*)d_in[10];
    const float* bhh     = (const float*)d_in[11];
    const float* out_W   = (const float*)d_in[12];
    const float* out_b   = (const float*)d_in[13];

    float* out = (float*)d_out;          // [0:V) logp, [V:V+H) h_new, [V+H:...) attn_w
    float* ws  = (float*)d_ws;           // ~76 KB scratch
    float* cat1        = ws;             // 2048
    float* attn_logits = ws + 2048;      // 512
    float* applied     = ws + 2560;      // 1024
    float* xbuf        = ws + 3584;      // 1024
    float* gi          = ws + 4608;      // 3072
    float* gh          = ws + 7680;      // 3072
    float* part        = ws + 10752;     // 8192
    float* pm          = ws + 18944;     // 64
    float* ps          = ws + 19008;     // 64
    float* lse         = ws + 19072;     // 1

    float* logp   = out;
    float* hnew   = out + V;
    float* attn_w = out + V + H;

    k_cat1        <<<8,   256, 0, stream>>>(tok, emb, hidden, cat1);
    k_attn_scores <<<64,  256, 0, stream>>>(cat1, attn_W, attn_b, attn_logits);
    k_softmax512  <<<1,   512, 0, stream>>>(attn_logits, attn_w);
    k_attn_partial<<<dim3(4, 8), 256, 0, stream>>>(attn_w, enc, part);
    k_attn_combine<<<4,   256, 0, stream>>>(part, applied);
    k_combine     <<<128, 256, 0, stream>>>(cat1, applied, comb_W, comb_b, xbuf);
    k_gru_gemv    <<<768, 256, 0, stream>>>(xbuf, hidden, Wih, Whh, bih, bhh, gi, gh);
    k_gru_update  <<<4,   256, 0, stream>>>(gi, gh, hidden, hnew);

    int nwaves  = (V + 15) / 16;                 // 3142
    int nblocks = (nwaves + 7) / 8;              // 393
    k_out_proj_wmma<<<nblocks, 256, 0, stream>>>(hnew, out_W, out_b, logp);

    k_lse_partial <<<NPB, 256, 0, stream>>>(logp, pm, ps);
    k_lse_final   <<<1,   NPB, 0, stream>>>(pm, ps, lse);
    k_lsm_apply   <<<NPB, 256, 0, stream>>>(logp, lse);
}